// MaxAttentionWords_16063177687232
// MI455X (gfx1250) — compile-verified
//
#include <hip/hip_runtime.h>
#include <stdint.h>

// Problem constants (match reference setup_inputs)
#define B_  32
#define S_  8192
#define D_  256
#define K_  64
#define TPB 256
#define EPT (S_ / TPB)        // 32 attention elements per thread
#define ROW_VEC (D_ / 4)      // 64 float4 per context row
#define TOT_VEC (K_ * ROW_VEC)// 4096 float4 of output per batch
#define PASS_VEC (TOT_VEC / 2)// 2048 float4 staged per pass (32 KB)
#define VEC_PER_THREAD (PASS_VEC / TPB) // 8 float4 per thread per pass

typedef __attribute__((address_space(3))) uint8_t lds_u8_t;

static __device__ __forceinline__ uint32_t lds_offset_of(const void* p) {
    // generic -> LDS addrspace, then truncate to the 32-bit LDS byte offset
    return (uint32_t)(uintptr_t)(lds_u8_t*)p;
}

static __device__ __forceinline__ uint32_t f32_orderable(float f) {
    uint32_t u = __float_as_uint(f);
    return (u & 0x80000000u) ? ~u : (u | 0x80000000u);
}

__global__ __launch_bounds__(TPB)
void topk_gather_kernel(const float* __restrict__ attention,
                        const float* __restrict__ context,
                        float* __restrict__ out)
{
    const int b    = blockIdx.x;
    const int t    = threadIdx.x;
    const int wave = t >> 5;
    const int lane = t & 31;

    __shared__ unsigned long long wmax[8];
    __shared__ unsigned long long winner;
    __shared__ uint32_t selrow[K_];
    __shared__ __align__(16) uint8_t stage[PASS_VEC * 16];  // 32 KB staging

    // ------------------------------------------------------------------
    // Phase 1: load this batch's 8192 scores into registers as sortable
    // 64-bit keys:  key = (orderable(f) << 32) | (S-1-idx)
    // Descending order over keys == jax.lax.top_k order (ties -> low idx).
    // ------------------------------------------------------------------
    unsigned long long keys[EPT];
    const float4* att4 = (const float4*)(attention + (size_t)b * S_);
    #pragma unroll
    for (int i = 0; i < EPT / 4; ++i) {
        const int vecIdx = i * TPB + t;        // 0..2047, coalesced
        const float4 v = att4[vecIdx];
        const uint32_t s0 = (uint32_t)vecIdx * 4u;
        keys[i*4+0] = ((unsigned long long)f32_orderable(v.x) << 32) | (unsigned long long)(S_ - 1u - (s0 + 0u));
        keys[i*4+1] = ((unsigned long long)f32_orderable(v.y) << 32) | (unsigned long long)(S_ - 1u - (s0 + 1u));
        keys[i*4+2] = ((unsigned long long)f32_orderable(v.z) << 32) | (unsigned long long)(S_ - 1u - (s0 + 2u));
        keys[i*4+3] = ((unsigned long long)f32_orderable(v.w) << 32) | (unsigned long long)(S_ - 1u - (s0 + 3u));
    }

    // ------------------------------------------------------------------
    // Phase 2: 64 rounds of argmax extraction. Keys are unique, so
    // "key < prev" exactly excludes everything already selected.
    // ------------------------------------------------------------------
    unsigned long long prev = ~0ull;
    for (int k = 0; k < K_; ++k) {
        unsigned long long best = 0ull;
        #pragma unroll
        for (int i = 0; i < EPT; ++i) {
            const unsigned long long key = keys[i];
            if (key < prev && key > best) best = key;
        }
        // wave32 max-reduce
        #pragma unroll
        for (int off = 16; off > 0; off >>= 1) {
            const unsigned long long o = __shfl_xor(best, off, 32);
            if (o > best) best = o;
        }
        if (lane == 0) wmax[wave] = best;
        __syncthreads();
        if (t == 0) {
            unsigned long long w = wmax[0];
            #pragma unroll
            for (int j = 1; j < 8; ++j) if (wmax[j] > w) w = wmax[j];
            winner    = w;
            selrow[k] = (uint32_t)(S_ - 1u - (uint32_t)(w & 0xffffffffu));
        }
        __syncthreads();
        prev = winner;
    }

    // ------------------------------------------------------------------
    // Phase 3: gather selected context rows -> out via the gfx1250 async
    // LDS path (global -> LDS -> global, tracked on ASYNCcnt). Each thread
    // only touches its own LDS slots, so per-wave s_wait_asynccnt is the
    // only sync needed; s_endpgm's implicit wait-idle drains final stores.
    // ------------------------------------------------------------------
    const size_t ctxBase = (size_t)b * S_ * D_;
    float* const outBase = out + (size_t)b * K_ * D_;
    const uint32_t stageBase = lds_offset_of(stage);

    for (int pass = 0; pass < 2; ++pass) {
        // WAR guard: previous pass's async stores must finish reading LDS
        asm volatile("s_wait_asynccnt 0x0" ::: "memory");

        #pragma unroll
        for (int i = 0; i < VEC_PER_THREAD; ++i) {
            const int slot = i * TPB + t;                 // 0..2047 in stage
            const int e    = pass * PASS_VEC + slot;      // 0..4095 global vec id
            const int row  = e >> 6;                      // /64 float4 per row
            const int col  = e & 63;
            const unsigned long long src =
                (unsigned long long)(uintptr_t)(context + ctxBase +
                    (size_t)selrow[row] * D_ + (size_t)col * 4);
            const uint32_t ldsAddr = stageBase + ((uint32_t)slot << 4);
            asm volatile("global_load_async_to_lds_b128 %0, %1, off"
                         :: "v"(ldsAddr), "v"(src) : "memory");
        }

        asm volatile("s_wait_asynccnt 0x0" ::: "memory");

        #pragma unroll
        for (int i = 0; i < VEC_PER_THREAD; ++i) {
            const int slot = i * TPB + t;
            const int e    = pass * PASS_VEC + slot;
            const unsigned long long dst =
                (unsigned long long)(uintptr_t)(outBase + (size_t)e * 4);
            const uint32_t ldsAddr = stageBase + ((uint32_t)slot << 4);
            asm volatile("global_store_async_from_lds_b128 %0, %1, off"
                         :: "v"(dst), "v"(ldsAddr) : "memory");
        }
    }
    // s_endpgm performs an implicit wait-idle -> async stores complete.
}

extern "C" void kernel_launch(void* const* d_in, const int* in_sizes, int n_in,
                              void* d_out, int out_size, void* d_ws, size_t ws_size,
                              hipStream_t stream) {
    (void)in_sizes; (void)n_in; (void)d_ws; (void)ws_size; (void)out_size;
    const float* attention = (const float*)d_in[0];   // [B, S] f32
    const float* context   = (const float*)d_in[1];   // [B, S, D] f32
    // d_in[2] is K (==64), baked in as a compile-time constant.
    float* out = (float*)d_out;                       // [B, K, D] f32

    topk_gather_kernel<<<B_, TPB, 0, stream>>>(attention, context, out);
}